// DepthFFN_mapping_12343736008723
// MI455X (gfx1250) — compile-verified
//
#include <hip/hip_runtime.h>
#include <stdint.h>

// ---------------------------------------------------------------------------
// CaDDN frustum-feature fusion for MI455X (gfx1250).
//   out[c,d,p] = feat[c,p] * softmax(10*onehot(idx(p)))[d]
// softmax collapses to two constants; kernel is store-bandwidth bound
// (~600 MB out / 23.3 TB/s ~= 26 us floor). Uses gfx1250 async global->LDS
// staging (ASYNCcnt) overlapped with LID index math, and NT store hints.
// ---------------------------------------------------------------------------

typedef float v2f __attribute__((ext_vector_type(2)));

#define NUM_BINS 80
#define DEPTH_MIN_F 2.0f
// bin_size = 2*(46.8-2.0)/(80*81) = 89.6/6480 ; INV_BIN = 1/bin_size
#define INV_BIN 72.32142857142857f
#define P_HOT 0.9963811493f   /* e^10/(e^10+80) */
#define P_COLD 4.5235634e-05f /* 1/(e^10+80)    */

__global__ __launch_bounds__(256) void frustum_fused_kernel(
    const float* __restrict__ md,    // [P] monodepth (H*W flattened)
    const float* __restrict__ feat,  // [C, P] image features
    float* __restrict__ out,         // [C, 80, P]
    int P)
{
    __shared__ float lds_feat[512];  // 2 KB staging buffer (256 lanes x 8B)

    const int c   = blockIdx.y;
    const int tid = threadIdx.x;
    const int pix = blockIdx.x * 512 + tid * 2;   // P is even -> whole pairs
    if (pix >= P) return;

    // --- 1) kick off async copy of this lane's 8 B of features into LDS ---
    // (gfx1250 GLOBAL_LOAD_ASYNC_TO_LDS, tracked by ASYNCcnt; LDS dest
    //  address is the low 32 bits of the flat shared pointer = LDS offset)
    const float* g = feat + (size_t)c * P + pix;
    unsigned loff = (unsigned)(uintptr_t)&lds_feat[2 * tid];
    asm volatile("global_load_async_to_lds_b64 %0, %1, off"
                 :: "v"(loff), "v"(g) : "memory");

    // --- 2) overlap: LID depth-bin index from monodepth (VALU sqrt) ---
    v2f m = *(const v2f*)(md + pix);
    float i0 = -0.5f + 0.5f * __builtin_sqrtf(1.0f + 8.0f * (m.x - DEPTH_MIN_F) * INV_BIN);
    float i1 = -0.5f + 0.5f * __builtin_sqrtf(1.0f + 8.0f * (m.y - DEPTH_MIN_F) * INV_BIN);
    // NaN (depth < DEPTH_MIN) or out-of-range -> bin 80 (dropped by softmax slice)
    int idx0 = (i0 >= 0.0f && i0 <= 80.0f) ? (int)i0 : NUM_BINS;
    int idx1 = (i1 >= 0.0f && i1 <= 80.0f) ? (int)i1 : NUM_BINS;

    // --- 3) wait for the async copy, read the staged features ---
#if __has_builtin(__builtin_amdgcn_s_wait_asynccnt)
    __builtin_amdgcn_s_wait_asynccnt(0);
#else
    asm volatile("s_wait_asynccnt 0x0" ::: "memory");
#endif
    v2f f  = *(const v2f*)&lds_feat[2 * tid];
    v2f fo = f * P_COLD;

    // --- 4) stream the 80 "cold" bins, non-temporal (write-once, > L2) ---
    float* obase = out + (size_t)c * NUM_BINS * P + pix;
    float* o = obase;
#pragma unroll 8
    for (int d = 0; d < NUM_BINS; ++d) {
        __builtin_nontemporal_store(fo, (v2f*)o);   // global_store_b64 ... nt
        o += P;
    }

    // --- 5) overwrite the hot bin (same wave: store-store order guaranteed) ---
    if (idx0 < NUM_BINS)
        __builtin_nontemporal_store(f.x * P_HOT, obase + (size_t)idx0 * P);
    if (idx1 < NUM_BINS)
        __builtin_nontemporal_store(f.y * P_HOT, obase + (size_t)idx1 * P + 1);
}

extern "C" void kernel_launch(void* const* d_in, const int* in_sizes, int n_in,
                              void* d_out, int out_size, void* d_ws, size_t ws_size,
                              hipStream_t stream) {
    (void)n_in; (void)out_size; (void)d_ws; (void)ws_size;
    const float* md   = (const float*)d_in[0];  // [1,94,311]
    const float* feat = (const float*)d_in[1];  // [1,64,94,311]
    float* out = (float*)d_out;                 // [1,64,80,94,311]

    const int P = in_sizes[0];            // 94*311 = 29234 (even)
    const int C = in_sizes[1] / P;        // 64

    dim3 grid((unsigned)((P / 2 + 255) / 256), (unsigned)C);
    frustum_fused_kernel<<<grid, 256, 0, stream>>>(md, feat, out, P);
}